// ModulatedDeformableConv2d_10033043604236
// MI455X (gfx1250) — compile-verified
//
#include <hip/hip_runtime.h>
#include <math.h>

typedef __attribute__((ext_vector_type(2))) float v2f;
typedef __attribute__((ext_vector_type(8))) float v8f;

#define BB 4
#define CC 64
#define HH 128
#define WW 128
#define OFFC 32
#define K2 9
#define CK 576          // CC * K2
#define SSTRIDE 577     // pad to avoid 64-bank conflicts (576 % 64 == 0)

// ---------------------------------------------------------------------------
// Kernel A: 3x3 offset/mask conv (27 out channels), pad=1, stride=1.
// One thread per (b, oc, h, w). Sigmoid applied to mask channels (oc >= 18).
// Output layout: om[((b*27 + oc)*128 + h)*128 + w]
// ---------------------------------------------------------------------------
__global__ __launch_bounds__(256) void offset_conv_kernel(
    const float* __restrict__ offset_feat,   // [B][32][128][128]
    const float* __restrict__ w_off,         // [27][32][3][3]
    const float* __restrict__ b_off,         // [27]
    float* __restrict__ om)                  // [B][27][128][128]
{
    int id = blockIdx.x * 256 + threadIdx.x;
    int w  = id & (WW - 1);
    int h  = (id >> 7) & (HH - 1);
    int oc = (id >> 14) % 27;
    int b  = id / (27 << 14);

    float acc = b_off[oc];
    const float* wbase = w_off + oc * (OFFC * 9);
    for (int ic = 0; ic < OFFC; ++ic) {
        const float* img = offset_feat + ((b * OFFC + ic) << 14);
        const float* wk  = wbase + ic * 9;
#pragma unroll
        for (int ky = 0; ky < 3; ++ky) {
            int y = h - 1 + ky;
            bool yok = (unsigned)y < (unsigned)HH;
#pragma unroll
            for (int kx = 0; kx < 3; ++kx) {
                int xw = w - 1 + kx;
                bool ok = yok && ((unsigned)xw < (unsigned)WW);
                float v = ok ? img[(y << 7) + xw] : 0.0f;
                acc = fmaf(v, wk[ky * 3 + kx], acc);
            }
        }
    }
    if (oc >= 18) acc = 1.0f / (1.0f + __expf(-acc));   // sigmoid for mask
    om[id] = acc;
}

// ---------------------------------------------------------------------------
// Kernel B: deformable sampling into LDS + fp32 WMMA GEMM.
// Workgroup = 128 threads (4 wave32). Tile = 16 pixels (one row strip) x 64
// output channels (16 channels per wave). Reduction depth CK=576 via
// 144 x V_WMMA_F32_16X16X4_F32.
// ---------------------------------------------------------------------------
__global__ __launch_bounds__(128) void deform_wmma_kernel(
    const float* __restrict__ x,        // [B][64][128][128]
    const float* __restrict__ om,       // [B][27][128][128] (dy/dx raw, mask sigmoided)
    const float* __restrict__ weight,   // [64][576]  (= [O][C][K2] flattened)
    const float* __restrict__ bias,     // [64]
    float* __restrict__ out)            // [B][64][128][128]
{
    __shared__ float ldsS[16 * SSTRIDE];   // sampled A-tile: S[p][c*9+k]
    __shared__ float ldsCo[144 * 6];       // per (p,k): y0, x0, 4 premul weights

    const int tid = threadIdx.x;
    const int t  = blockIdx.x;             // 4096 tiles = 4*128*8
    const int wo0 = (t & 7) << 4;
    const int ho  = (t >> 3) & (HH - 1);
    const int b   = t >> 10;

    // ---- Stage 0: bilinear coords for 16 pixels x 9 taps -------------------
    for (int i = tid; i < 144; i += 128) {
        int p = i / 9, k = i - p * 9;
        int pix   = (ho << 7) + wo0 + p;
        int obase = (b * 27) << 14;
        float dy = om[obase + ((2 * k)     << 14) + pix];
        float dx = om[obase + ((2 * k + 1) << 14) + pix];
        float m  = om[obase + ((18 + k)    << 14) + pix];
        float py = (float)(ho - 1 + (k / 3)) + dy;
        float px = (float)(wo0 + p - 1 + (k % 3)) + dx;
        float y0 = floorf(py), x0 = floorf(px);
        float wy = py - y0,    wx = px - x0;
        float* co = &ldsCo[i * 6];
        co[0] = y0;
        co[1] = x0;
        co[2] = (1.0f - wy) * (1.0f - wx) * m;
        co[3] = (1.0f - wy) * wx * m;
        co[4] = wy * (1.0f - wx) * m;
        co[5] = wy * wx * m;
    }
    __syncthreads();

    // ---- Stage 1: gather sampled values into LDS A-tile --------------------
    for (int idx = tid; idx < 16 * CK; idx += 128) {
        int k = idx % 9;
        int c = (idx / 9) & 63;
        int p = idx / CK;
        const float* co = &ldsCo[(p * 9 + k) * 6];
        int y0 = (int)co[0], x0 = (int)co[1];
        int y1 = y0 + 1,     x1 = x0 + 1;
        const float* img = x + ((b * CC + c) << 14);
        float v00 = ((unsigned)y0 < (unsigned)HH && (unsigned)x0 < (unsigned)WW) ? img[(y0 << 7) + x0] : 0.0f;
        float v01 = ((unsigned)y0 < (unsigned)HH && (unsigned)x1 < (unsigned)WW) ? img[(y0 << 7) + x1] : 0.0f;
        float v10 = ((unsigned)y1 < (unsigned)HH && (unsigned)x0 < (unsigned)WW) ? img[(y1 << 7) + x0] : 0.0f;
        float v11 = ((unsigned)y1 < (unsigned)HH && (unsigned)x1 < (unsigned)WW) ? img[(y1 << 7) + x1] : 0.0f;
        ldsS[p * SSTRIDE + c * 9 + k] =
            co[2] * v00 + co[3] * v01 + co[4] * v10 + co[5] * v11;
    }
    __syncthreads();

    // ---- Stage 2: fp32 WMMA GEMM: out[16p x 16o per wave] ------------------
    const int wave = tid >> 5;
    const int lane = tid & 31;
    const int mrow = lane & 15;            // A-matrix row (pixel)
    const int nn   = lane & 15;            // B/D column (out channel)
    const int kb   = (lane >> 4) << 1;     // K sub-pair per ISA 16x4 layout
    const int o    = (wave << 4) + nn;

    const float* arow = &ldsS[mrow * SSTRIDE + kb];
    const float* wrow = weight + o * CK + kb;

    v8f acc = {0.f, 0.f, 0.f, 0.f, 0.f, 0.f, 0.f, 0.f};
#pragma unroll 4
    for (int k0 = 0; k0 < CK; k0 += 4) {
        v2f a, bmat;
        a.x    = arow[k0];
        a.y    = arow[k0 + 1];
        bmat.x = wrow[k0];
        bmat.y = wrow[k0 + 1];
        acc = __builtin_amdgcn_wmma_f32_16x16x4_f32(
            /*neg_a=*/false, a, /*neg_b=*/false, bmat,
            /*c_mod=*/(short)0, acc, /*reuse_a=*/false, /*reuse_b=*/false);
    }

    // D layout: vgpr j, lane l -> m = j + 8*(l>>4), n = l&15 => contiguous wo
    const float bv = bias[o];
    const int mbase = (lane >> 4) << 3;
    float* obase = out + (((b * CC + o) << 7) + ho) * WW + wo0 + mbase;
    float4 lo = make_float4(acc[0] + bv, acc[1] + bv, acc[2] + bv, acc[3] + bv);
    float4 hi = make_float4(acc[4] + bv, acc[5] + bv, acc[6] + bv, acc[7] + bv);
    *(float4*)(obase)     = lo;
    *(float4*)(obase + 4) = hi;
}

// ---------------------------------------------------------------------------
extern "C" void kernel_launch(void* const* d_in, const int* in_sizes, int n_in,
                              void* d_out, int out_size, void* d_ws, size_t ws_size,
                              hipStream_t stream) {
    const float* x           = (const float*)d_in[0];
    const float* offset_feat = (const float*)d_in[1];
    const float* w_off       = (const float*)d_in[2];
    const float* b_off       = (const float*)d_in[3];
    const float* weight      = (const float*)d_in[4];
    const float* bias        = (const float*)d_in[5];
    float* out = (float*)d_out;
    float* om  = (float*)d_ws;   // [4][27][128][128] = 7.08 MB scratch

    // Kernel A: 4*27*128*128 = 1,769,472 threads
    offset_conv_kernel<<<6912, 256, 0, stream>>>(offset_feat, w_off, b_off, om);
    // Kernel B: 4096 tiles (16 pixels each) x 128 threads
    deform_wmma_kernel<<<4096, 128, 0, stream>>>(x, om, weight, bias, out);
}